// E3GNNConv_33844342292898
// MI455X (gfx1250) — compile-verified
//
#include <hip/hip_runtime.h>

// ---------------------------------------------------------------------------
// E3GNN conv for MI455X (gfx1250), wave32 + V_WMMA_F32_16X16X4_F32.
//
// Per 16-edge tile (one wave32), two phases over output-column halves:
//   acc0  = (s0*a0)   @ W00 + (1/sqrt3)*(sum_m s1_m*a1_m) @ W11  -> cols 0..63
//   accP  =  s0       @ W01                                       (p)
//   accQm = (s1_m*a0) @ W10  (m=0..2)
//   msg[64+3o+m] = p[o]*a1_m + accQm[o]
//   all scaled at scatter by f[e] = alpha * radial * (1/8) * (1/4)
//   (tail rows clamped; masked purely via f=0 at the atomic scatter)
//
// WMMA register layout assumptions (CDNA5 ISA 7.12.2):
//   A 16x4 f32 : lane l -> row l&15, pair {A[r][4k+2*(l>>4)], A[r][4k+2*(l>>4)+1]}
//   B 4x16 f32 : lane l -> col 16n+(l&15), pair {W[4k+2*(l>>4)][c], W[..+1][c]}
//                (pre-swizzled into workspace by e3_swizzle_kernel)
//   C/D 16x16  : v8f, element j of lane l -> row j+8*(l>>4), col l&15
//
// amdgpu_num_vgpr(256) caps allocation below the VGPR-MSB region: the
// pressure-aware scheduler throttles its software pipelining to fit, which
// removes the s_set_vgpr_msb + v_dual_mov juggling seen at 300+ VGPRs and
// restores occupancy for L2-latency hiding.
// ---------------------------------------------------------------------------

typedef float v2f __attribute__((ext_vector_type(2)));
typedef float v8f __attribute__((ext_vector_type(8)));

#define F_PER_NODE 256     // 4*MUL
#define NTILE_WAVES 4      // waves per block in the main kernel

__device__ __forceinline__ v8f wmma_f32_4(v2f a, v2f b, v8f c) {
  // 8-arg form: (neg_a, A, neg_b, B, c_mod, C, reuse_a, reuse_b)
  return __builtin_amdgcn_wmma_f32_16x16x4_f32(false, a, false, b, (short)0, c,
                                               false, false);
}

// --------------------------- zero the output -------------------------------
__global__ void e3_zero_kernel(float* __restrict__ p, long n) {
  long i = (long)blockIdx.x * blockDim.x + threadIdx.x;
  long stride = (long)gridDim.x * blockDim.x;
  for (; i < n; i += stride) p[i] = 0.0f;
}

// ------------------- pre-swizzle 4 weight matrices -------------------------
// wsB[((w*16 + k)*4 + n)*32 + lane] = { W[4k+2*(lane>>4)][16n+(lane&15)],
//                                       W[4k+2*(lane>>4)+1][16n+(lane&15)] }
__global__ void e3_swizzle_kernel(const float* __restrict__ w00,
                                  const float* __restrict__ w11,
                                  const float* __restrict__ w01,
                                  const float* __restrict__ w10,
                                  v2f* __restrict__ wsB) {
  int idx = blockIdx.x * blockDim.x + threadIdx.x;
  if (idx >= 4 * 16 * 4 * 32) return;
  int lane = idx & 31;
  int n = (idx >> 5) & 3;
  int k = (idx >> 7) & 15;
  int w = idx >> 11;
  const float* W = (w == 0) ? w00 : (w == 1) ? w11 : (w == 2) ? w01 : w10;
  int r = 4 * k + 2 * (lane >> 4);
  int c = 16 * n + (lane & 15);
  v2f v;
  v.x = W[r * 64 + c];
  v.y = W[(r + 1) * 64 + c];
  wsB[idx] = v;
}

// ------------------- radial MLP: f[e] = folded scalar ----------------------
__global__ __launch_bounds__(256)
void e3_rad_kernel(const float* __restrict__ emb, const float* __restrict__ w1,
                   const float* __restrict__ w2, float* __restrict__ fbuf,
                   int nE) {
  __shared__ float sw1[10 * 64];
  __shared__ float sw2[64];
  for (int i = threadIdx.x; i < 640; i += 256) sw1[i] = w1[i];
  if (threadIdx.x < 64) sw2[threadIdx.x] = w2[threadIdx.x];
  __syncthreads();
  int e = blockIdx.x * 256 + threadIdx.x;
  if (e >= nE) return;
  float em[10];
#pragma unroll
  for (int i = 0; i < 10; ++i) em[i] = emb[(size_t)e * 10 + i];
  const float rs10 = 0.316227766016838f;  // 1/sqrt(10)
  float racc = 0.0f;
#pragma unroll 8
  for (int j = 0; j < 64; ++j) {
    float z = 0.0f;
#pragma unroll
    for (int i = 0; i < 10; ++i) z = fmaf(em[i], sw1[i * 64 + j], z);
    z *= rs10;
    float h = (z / (1.0f + __expf(-z))) * 1.679177f;  // silu * SILU_NORM
    racc = fmaf(h, sw2[j], racc);
  }
  // fold: (1/sqrt(64)) * alpha(=1/sqrt(2*MUL)) * (1/sqrt(NUM_NEIGHBORS)=0.25)
  const float alpha = 0.0883883476483184f;  // 1/sqrt(128)
  fbuf[e] = racc * 0.125f * alpha * 0.25f;
}

// ----------------------------- main kernel ---------------------------------
__global__ __launch_bounds__(32 * NTILE_WAVES)
__attribute__((amdgpu_num_vgpr(256)))
void e3_conv_main(const float* __restrict__ node_in,
                  const float* __restrict__ edge_attr,
                  const int* __restrict__ edge_src,
                  const int* __restrict__ edge_dst,
                  const v2f* __restrict__ wsB,
                  const float* __restrict__ fbuf,
                  float* __restrict__ out, int n_edges, int n_tiles) {
  const int lane = threadIdx.x & 31;
  const int wave = threadIdx.x >> 5;
  const int tile = blockIdx.x * NTILE_WAVES + wave;
  if (tile >= n_tiles) return;  // wave-uniform: EXEC stays all-ones for WMMA

  const int e0 = tile * 16;
  const int row = lane & 15;
  const int half = lane >> 4;

  int e = e0 + row;
  if (e >= n_edges) e = n_edges - 1;  // clamp; masked via f=0 at scatter

  const int src = edge_src[e];
  const float* __restrict__ srcp = node_in + (size_t)src * F_PER_NODE;
  const float inv3 = 0.577350269189626f;  // 1/sqrt(3)
  const float4* __restrict__ attr4 = (const float4*)edge_attr;

  // Two phases over the output-column halves (n = 2*ph + t, t in {0,1}).
  // Accumulator pressure: 5*2 v8f = 80 VGPRs per phase.
#pragma clang loop unroll(disable)
  for (int ph = 0; ph < 2; ++ph) {
    // Stop LICM from hoisting the phase-invariant A-fragment loads/math out
    // of this loop (keeps live VGPRs per phase bounded).
    asm volatile("" ::: "memory");

    const float4 at = attr4[e];
    const float a0 = at.x, a1x = at.y, a1y = at.z, a1z = at.w;

    v8f c0[2], cP[2], cQ0[2], cQ1[2], cQ2[2];
#pragma unroll
    for (int t = 0; t < 2; ++t) {
      v8f z = {0.f, 0.f, 0.f, 0.f, 0.f, 0.f, 0.f, 0.f};
      c0[t] = z; cP[t] = z; cQ0[t] = z; cQ1[t] = z; cQ2[t] = z;
    }

#pragma unroll
    for (int k = 0; k < 16; ++k) {
      const int i0 = 4 * k + 2 * half;  // this lane's K pair: i0, i0+1
      v2f s0p = *(const v2f*)(srcp + i0);           // 8B aligned (i0 even)
      const float* s1p = srcp + 64 + 3 * i0;        // 8B aligned
      v2f u0 = *(const v2f*)(s1p);       // s1[i0][0],   s1[i0][1]
      v2f u1 = *(const v2f*)(s1p + 2);   // s1[i0][2],   s1[i0+1][0]
      v2f u2 = *(const v2f*)(s1p + 4);   // s1[i0+1][1], s1[i0+1][2]

      v2f aS0A0; aS0A0.x = s0p.x * a0;  aS0A0.y = s0p.y * a0;
      v2f aT1;
      aT1.x = inv3 * (u0.x * a1x + u0.y * a1y + u1.x * a1z);
      aT1.y = inv3 * (u1.y * a1x + u2.x * a1y + u2.y * a1z);
      v2f aQ0; aQ0.x = u0.x * a0; aQ0.y = u1.y * a0;
      v2f aQ1; aQ1.x = u0.y * a0; aQ1.y = u2.x * a0;
      v2f aQ2; aQ2.x = u1.x * a0; aQ2.y = u2.y * a0;

#pragma unroll
      for (int t = 0; t < 2; ++t) {
        const int n = ph * 2 + t;
        v2f b00 = wsB[((0 * 16 + k) * 4 + n) * 32 + lane];
        v2f b11 = wsB[((1 * 16 + k) * 4 + n) * 32 + lane];
        v2f b01 = wsB[((2 * 16 + k) * 4 + n) * 32 + lane];
        v2f b10 = wsB[((3 * 16 + k) * 4 + n) * 32 + lane];
        c0[t]  = wmma_f32_4(aS0A0, b00, c0[t]);
        c0[t]  = wmma_f32_4(aT1,   b11, c0[t]);
        cP[t]  = wmma_f32_4(s0p,   b01, cP[t]);
        cQ0[t] = wmma_f32_4(aQ0,   b10, cQ0[t]);
        cQ1[t] = wmma_f32_4(aQ1,   b10, cQ1[t]);
        cQ2[t] = wmma_f32_4(aQ2,   b10, cQ2[t]);
      }
    }

    // ---------------- scatter epilogue for this column half ----------------
#pragma unroll
    for (int j = 0; j < 8; ++j) {
      const int r = j + 8 * half;  // tile row of this lane's j-th acc element
      int er = e0 + r;
      const bool v = (er < n_edges);
      if (!v) er = n_edges - 1;
      float fr = fbuf[er];
      if (!v) fr = 0.0f;
      const int dr = edge_dst[er];
      const float4 ar = attr4[er];
      float* __restrict__ orow = out + (size_t)dr * F_PER_NODE;
#pragma unroll
      for (int t = 0; t < 2; ++t) {
        const int o = (ph * 2 + t) * 16 + row;
        unsafeAtomicAdd(orow + o, c0[t][j] * fr);
        const float p = cP[t][j];
        unsafeAtomicAdd(orow + 64 + 3 * o + 0, (p * ar.y + cQ0[t][j]) * fr);
        unsafeAtomicAdd(orow + 64 + 3 * o + 1, (p * ar.z + cQ1[t][j]) * fr);
        unsafeAtomicAdd(orow + 64 + 3 * o + 2, (p * ar.w + cQ2[t][j]) * fr);
      }
    }
  }
}

// ----------------------------- launcher ------------------------------------
extern "C" void kernel_launch(void* const* d_in, const int* in_sizes, int n_in,
                              void* d_out, int out_size, void* d_ws,
                              size_t ws_size, hipStream_t stream) {
  const float* node_in   = (const float*)d_in[0];
  const float* edge_attr = (const float*)d_in[1];
  const float* edge_len  = (const float*)d_in[2];
  const float* w00 = (const float*)d_in[3];
  const float* w11 = (const float*)d_in[4];
  const float* w01 = (const float*)d_in[5];
  const float* w10 = (const float*)d_in[6];
  const float* fc1 = (const float*)d_in[7];
  const float* fc2 = (const float*)d_in[8];
  const int* esrc = (const int*)d_in[9];
  const int* edst = (const int*)d_in[10];
  const int n_edges = in_sizes[9];
  float* out = (float*)d_out;

  // workspace layout: [0, 64KB) swizzled B fragments; then f[e] per edge.
  v2f* wsB = (v2f*)d_ws;
  float* fbuf = (float*)((char*)d_ws + (size_t)(4 * 16 * 4 * 32) * sizeof(v2f));

  e3_zero_kernel<<<1024, 256, 0, stream>>>(out, (long)out_size);
  e3_swizzle_kernel<<<(4 * 16 * 4 * 32 + 255) / 256, 256, 0, stream>>>(
      w00, w11, w01, w10, wsB);
  e3_rad_kernel<<<(n_edges + 255) / 256, 256, 0, stream>>>(edge_len, fc1, fc2,
                                                           fbuf, n_edges);

  const int n_tiles = (n_edges + 15) / 16;
  const int blocks = (n_tiles + NTILE_WAVES - 1) / NTILE_WAVES;
  e3_conv_main<<<blocks, 32 * NTILE_WAVES, 0, stream>>>(
      node_in, edge_attr, esrc, edst, wsB, fbuf, out, n_edges, n_tiles);
}